// ModalSynergyCrossWeight_81698867904876
// MI455X (gfx1250) — compile-verified
//
#include <hip/hip_runtime.h>
#include <hip/hip_bf16.h>

typedef __attribute__((ext_vector_type(2))) float v2f;
typedef __attribute__((ext_vector_type(8))) float v8f;

// One wave (32 lanes) computes a 16x16 tile of dist2 = l2 + r2 - 2*(L . R^T)
// per iteration using V_WMMA_F32_16X16X4_F32 (K=3 padded to 4), tracking a
// running argmin per output row. Radar {x,y,z,r2} staged in dynamic LDS.
__global__ void ModalSynergyCrossWeight_kernel(const float* __restrict__ feat,   // [nRadar,64]
                                               const float* __restrict__ lidar,  // [nLidar,3]
                                               const float* __restrict__ radar,  // [nRadar,3]
                                               float* __restrict__ out,          // [nLidar]
                                               int nRadar, int nLidar) {
    extern __shared__ float smem[]; // nRadar x {x,y,z,r2}

    // Stage radar coords + precomputed r2 into LDS (whole block cooperates).
    for (int i = threadIdx.x; i < nRadar; i += blockDim.x) {
        float x = radar[3 * i + 0];
        float y = radar[3 * i + 1];
        float z = radar[3 * i + 2];
        smem[4 * i + 0] = x;
        smem[4 * i + 1] = y;
        smem[4 * i + 2] = z;
        smem[4 * i + 3] = fmaf(x, x, fmaf(y, y, z * z));
    }
    __syncthreads();

    const int wave  = threadIdx.x >> 5;
    const int lane  = threadIdx.x & 31;
    const int l16   = lane & 15;
    const int hi    = lane >> 4;          // 0 for lanes 0-15, 1 for lanes 16-31
    const int mBase = (blockIdx.x * (blockDim.x >> 5) + wave) * 16;
    if (mBase >= nLidar) return;          // wave-uniform: EXEC stays all-ones

    // ---- Load A matrix: lidar tile, 16x4 f32 (K=3 padded with 0) ----
    // ISA layout: lanes 0-15 VGPR0=K0,VGPR1=K1 ; lanes 16-31 VGPR0=K2,VGPR1=K3.
    int row  = mBase + l16;
    int crow = row < nLidar ? row : nLidar - 1;   // clamp, keep EXEC full
    float cx = lidar[3 * crow + 0];
    float cy = lidar[3 * crow + 1];
    float cz = lidar[3 * crow + 2];
    float l2row = fmaf(cx, cx, fmaf(cy, cy, cz * cz));
    v2f a;
    a.x = hi ? cz : cx;
    a.y = hi ? 0.0f : cy;

    // l2 needed per accumulator slot v: row (v + 8*hi); lane (v+8*hi) holds it.
    float l2s[8];
#pragma unroll
    for (int v = 0; v < 8; ++v)
        l2s[v] = __shfl(l2row, v + 8 * hi, 32);

    float best[8];
    int   bidx[8];
#pragma unroll
    for (int v = 0; v < 8; ++v) { best[v] = 3.0e38f; bidx[v] = 0; }

    const int nTiles = (nRadar + 15) >> 4;
    for (int nb = 0; nb < nTiles; ++nb) {
        int  n     = (nb << 4) + l16;
        bool valid = n < nRadar;
        int  cn    = valid ? n : 0;

        const float4 p = ((const float4*)smem)[cn];   // ds_load_b128 {x,y,z,r2}
        // B matrix (radar^T, 4x16): mirror of A layout with M<->N.
        v2f b;
        b.x = hi ? p.z : p.x;
        b.y = hi ? 0.0f : p.y;
        float r2 = valid ? p.w : 3.0e38f;             // poison OOB columns

        v8f c = {};
        // D = A x B : cross[m][n] tile. 8 args: (neg_a, A, neg_b, B, c_mod, C, reuse_a, reuse_b)
        c = __builtin_amdgcn_wmma_f32_16x16x4_f32(false, a, false, b,
                                                  (short)0, c, false, false);

#pragma unroll
        for (int v = 0; v < 8; ++v) {
            float dist = fmaf(-2.0f, c[v], l2s[v] + r2);
            if (dist < best[v]) { best[v] = dist; bidx[v] = n; } // strict < keeps first idx
        }
    }

    // Reduce min/argmin across the 16 lanes of each half-wave (xor masks < 16
    // stay inside the half). Tie-break on lower index to match jnp.argmin.
#pragma unroll
    for (int v = 0; v < 8; ++v) {
#pragma unroll
        for (int off = 8; off >= 1; off >>= 1) {
            float ob = __shfl_xor(best[v], off, 32);
            int   oi = __shfl_xor(bidx[v], off, 32);
            if (ob < best[v] || (ob == best[v] && oi < bidx[v])) {
                best[v] = ob;
                bidx[v] = oi;
            }
        }
        int orow = mBase + v + 8 * hi;   // lane v of each half owns one row
        if (l16 == v && orow < nLidar) {
            int   j   = bidx[v];
            float dop = fabsf(feat[j * 64 + 0]);
            float pw  = feat[j * 64 + 1];
            float s   = 0.6f * dop + 0.4f * pw;
            out[orow] = 1.0f / (1.0f + __expf(-s));
        }
    }
}

extern "C" void kernel_launch(void* const* d_in, const int* in_sizes, int n_in,
                              void* d_out, int out_size, void* d_ws, size_t ws_size,
                              hipStream_t stream) {
    const float* feat  = (const float*)d_in[0];   // [nRadar, 64]
    const float* lidar = (const float*)d_in[1];   // [nLidar, 3]
    const float* radar = (const float*)d_in[2];   // [nRadar, 3]
    float*       out   = (float*)d_out;           // [nLidar]

    const int nLidar = in_sizes[1] / 3;           // 50000
    const int nRadar = in_sizes[2] / 3;           // 5000

    const int wavesPerBlock = 8;                  // 256 threads = 8 wave32
    const int mTiles = (nLidar + 15) / 16;        // 3125 row-tiles
    const int blocks = (mTiles + wavesPerBlock - 1) / wavesPerBlock;
    const size_t smemBytes = (size_t)nRadar * 4 * sizeof(float);  // 80 KB <= 320 KB/WGP

    ModalSynergyCrossWeight_kernel<<<blocks, wavesPerBlock * 32, smemBytes, stream>>>(
        feat, lidar, radar, out, nRadar, nLidar);
}